// LoopyBeliefPropagation_48576080117802
// MI455X (gfx1250) — compile-verified
//
#include <hip/hip_runtime.h>

// Problem constants (B=4, L=128, MAX_ITER=3 per the reference).
#define LL 128
#define BB 4
#define NITER 3
#define PLANE (LL * LL * LL)          // 2,097,152 floats per (b,c) plane
#define MBUF_FLOATS (8 * PLANE)       // 16,777,216 floats per message buffer (64 MiB)

typedef __attribute__((ext_vector_type(2))) float v2f;
typedef __attribute__((ext_vector_type(8))) float v8f;

__device__ __forceinline__ float laexp(float a, float b) {
  // numerically stable logaddexp
  float mx = fmaxf(a, b);
  float mn = fminf(a, b);
  return mx + log1pf(__expf(mn - mx));
}

// ---------------------------------------------------------------------------
// Kernel A: bt[x][y][b][c] = p_edge[x,y,b,c] + 3 * sum_z m[b*2+c][x][y][z] * w
//   w(x,y,z,b) = mask[b,z,x]*mask[b,y,x]*(z != min(x,y))*(z != max(x,y))
// Masked row-sum over z expressed as (M .* W) x ones via
// V_WMMA_F32_16X16X4_F32. Both channels (c=0,1) share the weight w(z) and are
// interleaved in one z-pass (4 accumulator chains). Weight gating is a
// branchless v_cndmask; smask pairs load as one ds_load_b64.
// Block = 128 threads = 4 waves; wave w handles batch b = w.
// Grid = (L, L/16): one 16-row y-tile per block per x.
// ---------------------------------------------------------------------------
__global__ void lbp_reduce(const float* __restrict__ mcur,
                           float* __restrict__ bt,
                           const float* __restrict__ s_edge,
                           const unsigned char* __restrict__ mask,
                           int first) {
  const int x    = blockIdx.x;
  const int Y0   = blockIdx.y * 16;
  const int lane = threadIdx.x & 31;
  const int b    = threadIdx.x >> 5;  // wave id == batch index

  if (first) {
    // messages are zero: bt = p_edge = s_edge[b, y, x, c]
    const int y = Y0 + (lane & 15);
    const int c = lane >> 4;
    bt[((x * LL + y) * BB + b) * 2 + c] = s_edge[((b * LL + y) * LL + x) * 2 + c];
    return;
  }

  __shared__ float smask[BB][LL];  // smask[b][k] = mask[b, k, x] as 0/1 float
  for (int z = lane; z < LL; z += 32)
    smask[b][z] = mask[(b * LL + z) * LL + x] ? 1.0f : 0.0f;
  __syncthreads();

  // WMMA 16x16x4 f32 A-fragment layout: lanes 0-15 -> M=lane, K=0,1 in v0,v1;
  // lanes 16-31 -> M=lane-16, K=2,3.
  const int   row = lane & 15;
  const int   kh  = lane >> 4;      // 0 or 1 -> z offset 0 or 2
  const int   y   = Y0 + row;
  const float fy  = smask[b][y];    // mask[b, y, x]
  const int   h   = (x < y) ? x : y;
  const int   d   = (x < y) ? y : x;

  const float* mp0 = mcur + (size_t)((b * 2 + 0) * LL + x) * (LL * LL) + (size_t)y * LL;
  const float* mp1 = mp0 + (size_t)PLANE;  // c=1 plane, same (b, x, y)

  v2f ones;
  ones.x = 1.0f;
  ones.y = 1.0f;

  v8f a0c0 = {}, a1c0 = {};  // two chains for c=0
  v8f a0c1 = {}, a1c1 = {};  // two chains for c=1

  for (int z0 = 0; z0 < LL; z0 += 8) {
    {
      const int z = z0 + 2 * kh;
      const float s0 = smask[b][z];      // adjacent pair -> ds_load_b64
      const float s1 = smask[b][z + 1];
      const float g0 = ((z     != h) && (z     != d)) ? fy : 0.0f;  // v_cndmask
      const float g1 = ((z + 1 != h) && (z + 1 != d)) ? fy : 0.0f;
      v2f w;
      w.x = g0 * s0;
      w.y = g1 * s1;
      v2f mv0 = *(const v2f*)(mp0 + z);
      v2f mv1 = *(const v2f*)(mp1 + z);
      v2f f0, f1;
      f0.x = mv0.x * w.x; f0.y = mv0.y * w.y;
      f1.x = mv1.x * w.x; f1.y = mv1.y * w.y;
      a0c0 = __builtin_amdgcn_wmma_f32_16x16x4_f32(false, f0, false, ones,
                                                   (short)0, a0c0, false, false);
      a0c1 = __builtin_amdgcn_wmma_f32_16x16x4_f32(false, f1, false, ones,
                                                   (short)0, a0c1, false, false);
    }
    {
      const int z = z0 + 4 + 2 * kh;
      const float s0 = smask[b][z];
      const float s1 = smask[b][z + 1];
      const float g0 = ((z     != h) && (z     != d)) ? fy : 0.0f;
      const float g1 = ((z + 1 != h) && (z + 1 != d)) ? fy : 0.0f;
      v2f w;
      w.x = g0 * s0;
      w.y = g1 * s1;
      v2f mv0 = *(const v2f*)(mp0 + z);
      v2f mv1 = *(const v2f*)(mp1 + z);
      v2f f0, f1;
      f0.x = mv0.x * w.x; f0.y = mv0.y * w.y;
      f1.x = mv1.x * w.x; f1.y = mv1.y * w.y;
      a1c0 = __builtin_amdgcn_wmma_f32_16x16x4_f32(false, f0, false, ones,
                                                   (short)0, a1c0, false, false);
      a1c1 = __builtin_amdgcn_wmma_f32_16x16x4_f32(false, f1, false, ones,
                                                   (short)0, a1c1, false, false);
    }
  }
  v8f accc0 = a0c0 + a1c0;
  v8f accc1 = a0c1 + a1c1;

  // D fragment: lane 0 holds D[0..7, 0] in v0..7, lane 16 holds D[8..15, 0].
  if (lane == 0 || lane == 16) {
    const int rb = (lane == 0) ? 0 : 8;
#pragma unroll
    for (int r = 0; r < 8; ++r) {
      const int yy = Y0 + rb + r;
      bt[((x * LL + yy) * BB + b) * 2 + 0] =
          s_edge[((b * LL + yy) * LL + x) * 2 + 0] + 3.0f * accc0[r];
      bt[((x * LL + yy) * BB + b) * 2 + 1] =
          s_edge[((b * LL + yy) * LL + x) * 2 + 1] + 3.0f * accc1[r];
    }
  }
}

// ---------------------------------------------------------------------------
// Kernel B: message update + y<->z transpose.
//   mm0 = bt0[x,y,b] - m0[x,y,z,b]; mm1 = bt1 - m1
//   n0 = logaddexp(mm0, mm1); n1 = logaddexp(mm0, mm1 + p_sib[x,y,z,b])
//   l  = logaddexp(n0, n1)
//   m_new[x, z, y, b, c] = n_c - l            (note the y<->z swap on write)
// Grid = (L, L/32, L/32) tiles; block (32, 8); batches looped inside.
// Transpose staged through padded LDS for fully coalesced global writes.
// ---------------------------------------------------------------------------
__global__ void lbp_update(const float* __restrict__ mcur,
                           float* __restrict__ mnxt,
                           const float* __restrict__ bt,
                           const float* __restrict__ s_sib,
                           int first) {
  const int x  = blockIdx.x;
  const int Y0 = blockIdx.y * 32;
  const int Z0 = blockIdx.z * 32;
  const int tx = threadIdx.x;  // 0..31
  const int ty = threadIdx.y;  // 0..7

  __shared__ float sm0[32][33];
  __shared__ float sm1[32][33];
  __shared__ float sp[32][33];
  __shared__ float so0[32][33];
  __shared__ float so1[32][33];
  __shared__ float sbt0[32];
  __shared__ float sbt1[32];

  for (int b = 0; b < BB; ++b) {
    // ---- load phase (coalesced in z) ----
#pragma unroll
    for (int r = 0; r < 4; ++r) {
      const int yl = ty + 8 * r;
      const int y  = Y0 + yl;
      const int z  = Z0 + tx;
      sp[yl][tx] = s_sib[((b * LL + y) * LL + x) * LL + z];
      if (first) {
        sm0[yl][tx] = 0.0f;
        sm1[yl][tx] = 0.0f;
      } else {
        sm0[yl][tx] = mcur[(((b * 2 + 0) * LL + x) * LL + y) * LL + z];
        sm1[yl][tx] = mcur[(((b * 2 + 1) * LL + x) * LL + y) * LL + z];
      }
      if (r == 0 && b < BB - 1) {
        // warm L2 for the next batch's s_sib tile (global_prefetch_b8)
        __builtin_prefetch(&s_sib[(((b + 1) * LL + y) * LL + x) * LL + z], 0, 0);
      }
    }
    if (ty == 0) {
      const int y = Y0 + tx;
      sbt0[tx] = bt[((x * LL + y) * BB + b) * 2 + 0];
      sbt1[tx] = bt[((x * LL + y) * BB + b) * 2 + 1];
    }
    __syncthreads();

    // ---- compute phase ----
#pragma unroll
    for (int r = 0; r < 4; ++r) {
      const int yl = ty + 8 * r;
      const int zl = tx;
      const float m0  = sm0[yl][zl];
      const float m1  = sm1[yl][zl];
      const float ps  = sp[yl][zl];
      const float mm0 = sbt0[yl] - m0;
      const float mm1 = sbt1[yl] - m1;
      const float n0  = laexp(mm0, mm1);
      const float n1  = laexp(mm0, mm1 + ps);
      const float l   = laexp(n0, n1);
      so0[zl][yl] = n0 - l;  // padded stride 33 -> conflict-free transpose
      so1[zl][yl] = n1 - l;
    }
    __syncthreads();

    // ---- write phase (coalesced in y) ----
#pragma unroll
    for (int r = 0; r < 4; ++r) {
      const int zl = ty + 8 * r;
      const int z  = Z0 + zl;
      const int yy = Y0 + tx;
      mnxt[(((b * 2 + 0) * LL + x) * LL + z) * LL + yy] = so0[zl][tx];
      mnxt[(((b * 2 + 1) * LL + x) * LL + z) * LL + yy] = so1[zl][tx];
    }
    __syncthreads();
  }
}

// ---------------------------------------------------------------------------
// Kernel C: out[b, y, x, c] = softmax_c(bt[x, y, b, :])
// ---------------------------------------------------------------------------
__global__ void lbp_final(const float* __restrict__ bt, float* __restrict__ out) {
  const int t = blockIdx.x * blockDim.x + threadIdx.x;  // t = (b*L + y)*L + x
  if (t >= BB * LL * LL) return;
  const int x = t % LL;
  const int y = (t / LL) % LL;
  const int b = t / (LL * LL);
  const float b0 = bt[((x * LL + y) * BB + b) * 2 + 0];
  const float b1 = bt[((x * LL + y) * BB + b) * 2 + 1];
  const float mx  = fmaxf(b0, b1);
  const float e0  = __expf(b0 - mx);
  const float e1  = __expf(b1 - mx);
  const float inv = 1.0f / (e0 + e1);
  out[t * 2 + 0] = e0 * inv;
  out[t * 2 + 1] = e1 * inv;
}

extern "C" void kernel_launch(void* const* d_in, const int* in_sizes, int n_in,
                              void* d_out, int out_size, void* d_ws, size_t ws_size,
                              hipStream_t stream) {
  (void)in_sizes; (void)n_in; (void)out_size; (void)ws_size;

  const float*         s_edge = (const float*)d_in[0];
  const float*         s_sib  = (const float*)d_in[1];
  // d_in[2] (s_cop) and d_in[3] (s_grd) are never read by the reference math.
  const unsigned char* mask   = (const unsigned char*)d_in[4];
  float*               out    = (float*)d_out;

  // Workspace: two 64 MiB message buffers (double-buffered y<->z transpose)
  // plus bt[L][L][B][2] (512 KiB).
  float* buf0 = (float*)d_ws;
  float* buf1 = buf0 + MBUF_FLOATS;
  float* bt   = buf1 + MBUF_FLOATS;
  float* bufs[2] = {buf0, buf1};

  const dim3 gridA(LL, LL / 16);
  const dim3 blockA(128);
  const dim3 gridB(LL, LL / 32, LL / 32);
  const dim3 blockB(32, 8);

  int cur = 1, nxt = 0;  // iteration 0 reads nothing (first flag), writes buf0
  for (int it = 0; it < NITER; ++it) {
    const int first = (it == 0) ? 1 : 0;
    lbp_reduce<<<gridA, blockA, 0, stream>>>(bufs[cur], bt, s_edge, mask, first);
    lbp_update<<<gridB, blockB, 0, stream>>>(bufs[cur], bufs[nxt], bt, s_sib, first);
    cur = nxt;
    nxt ^= 1;
  }
  lbp_reduce<<<gridA, blockA, 0, stream>>>(bufs[cur], bt, s_edge, mask, 0);
  lbp_final<<<(BB * LL * LL + 255) / 256, 256, 0, stream>>>(bt, out);
}